// PointTransformerBlock_83442624627329
// MI455X (gfx1250) — compile-verified
//
#include <hip/hip_runtime.h>
#include <math.h>

// ---------------- problem constants ----------------
#define B_    4
#define NP_   8192
#define CIN_  32
#define COUT_ 64
#define KNN_  16
#define MP_   2048          // 0.25 * NP
#define R2_   0.01f         // 0.1^2

typedef float v2f __attribute__((ext_vector_type(2)));
typedef float v8f __attribute__((ext_vector_type(8)));

// D = A(16x4 f32) * B(4x16 f32) + C  -- CDNA5 V_WMMA_F32_16X16X4_F32
__device__ __forceinline__ v8f wmma4(v2f a, v2f b, v8f c) {
  return __builtin_amdgcn_wmma_f32_16x16x4_f32(
      /*neg_a=*/false, a, /*neg_b=*/false, b,
      /*c_mod=*/(short)0, c, /*reuse_a=*/false, /*reuse_b=*/false);
}

// =====================================================================
// Generic C[M x 64] = A[M x K] @ W[K x 64] (+bias), WMMA f32 16x16x4.
// grid.x = M/16 row blocks; 128 threads = 4 waves; wave w -> N tile 16*w.
// =====================================================================
__global__ __launch_bounds__(128) void gemm_n64_kernel(
    const float* __restrict__ A, const float* __restrict__ W,
    const float* __restrict__ bias, float* __restrict__ C, int K)
{
  const int wv   = threadIdx.x >> 5;
  const int lane = threadIdx.x & 31;
  const int m0   = blockIdx.x * 16;
  const int n0   = wv * 16;
  const int n    = lane & 15;        // column within tile / row within A block
  const int half = lane >> 4;

  v8f acc = {};
  for (int k0 = 0; k0 < K; k0 += 4) {
    v2f a, b;
    const float* arow = A + (size_t)(m0 + n) * K + k0 + 2 * half;
    a.x = arow[0];
    a.y = arow[1];
    b.x = W[(k0 + 2 * half) * 64 + n0 + n];
    b.y = W[(k0 + 2 * half + 1) * 64 + n0 + n];
    acc = wmma4(a, b, acc);
  }
  const float bn = bias ? bias[n0 + n] : 0.0f;
#pragma unroll
  for (int r = 0; r < 8; ++r) {
    const int row = r + 8 * half;
    C[(size_t)(m0 + row) * 64 + n0 + n] = acc[r] + bn;
  }
}

// =====================================================================
// Farthest point sampling: one 1024-thread block per cloud.
// min_d lives in LDS (32 KB); per-step argmax via wave shfl + LDS tree.
// First-occurrence tie-break (lowest index wins) matches jnp.argmax.
// =====================================================================
__global__ __launch_bounds__(1024) void fps_kernel(
    const float* __restrict__ pos, int* __restrict__ idx_out)
{
  const int b = blockIdx.x;
  const float* p = pos + (size_t)b * NP_ * 3;
  __shared__ float mind[NP_];
  __shared__ float red_v[32];
  __shared__ int   red_i[32];
  __shared__ int   s_last;
  const int t = threadIdx.x;

  for (int i = t; i < NP_; i += 1024) mind[i] = 3.4e38f;
  if (t == 0) { idx_out[b * MP_] = 0; s_last = 0; }
  __syncthreads();

  int last = 0;
  for (int s = 1; s < MP_; ++s) {
    const float lx = p[last * 3 + 0];
    const float ly = p[last * 3 + 1];
    const float lz = p[last * 3 + 2];

    float bestv = -1.0f; int besti = NP_;
    const int i0 = t * 8;                      // contiguous ownership
#pragma unroll
    for (int k = 0; k < 8; ++k) {
      const int i = i0 + k;
      const float dx = p[i * 3 + 0] - lx;
      const float dy = p[i * 3 + 1] - ly;
      const float dz = p[i * 3 + 2] - lz;
      const float d  = dx * dx + dy * dy + dz * dz;
      const float md = fminf(mind[i], d);
      mind[i] = md;
      if (md > bestv) { bestv = md; besti = i; }     // strict > keeps lowest idx
    }
    for (int off = 1; off < 32; off <<= 1) {
      const float ov = __shfl_xor(bestv, off, 32);
      const int   oi = __shfl_xor(besti, off, 32);
      if (ov > bestv || (ov == bestv && oi < besti)) { bestv = ov; besti = oi; }
    }
    if ((t & 31) == 0) { red_v[t >> 5] = bestv; red_i[t >> 5] = besti; }
    __syncthreads();
    if (t < 32) {
      bestv = red_v[t]; besti = red_i[t];
      for (int off = 1; off < 32; off <<= 1) {
        const float ov = __shfl_xor(bestv, off, 32);
        const int   oi = __shfl_xor(besti, off, 32);
        if (ov > bestv || (ov == bestv && oi < besti)) { bestv = ov; besti = oi; }
      }
      if (t == 0) { s_last = besti; idx_out[b * MP_ + s] = besti; }
    }
    __syncthreads();
    last = s_last;
  }
}

// =====================================================================
// Gather sampled points: pos_s/batch_s (outputs), x_s (workspace).
// One thread per (sample, channel).
// =====================================================================
__global__ __launch_bounds__(256) void gather_kernel(
    const float* __restrict__ pos, const float* __restrict__ x_proj,
    const int* __restrict__ fidx,
    float* __restrict__ pos_s, float* __restrict__ x_s, int* __restrict__ batch_s)
{
  const int gid = blockIdx.x * blockDim.x + threadIdx.x;
  const int i = gid >> 6;
  const int c = gid & 63;
  if (i >= B_ * MP_) return;
  const int b = i / MP_;
  const int src = b * NP_ + fidx[i];
  x_s[(size_t)i * 64 + c] = x_proj[(size_t)src * 64 + c];
  if (c < 3) pos_s[i * 3 + c] = pos[(size_t)src * 3 + c];
  if (c == 0) batch_s[i] = b;
}

// =====================================================================
// Radius-kNN: one wave per sample. Per-lane sorted top-16 insertion over
// 256 candidates, then 16 rounds of wave arg-min merge (ties -> lowest j).
// Invalid (fewer than K in radius) -> -1.
// =====================================================================
__global__ __launch_bounds__(128) void knn_kernel(
    const float* __restrict__ pos, const float* __restrict__ pos_s,
    int* __restrict__ nbr)
{
  const int wave = blockIdx.x * (blockDim.x >> 5) + (threadIdx.x >> 5);
  const int lane = threadIdx.x & 31;
  if (wave >= B_ * MP_) return;
  const int b = wave / MP_;
  const float* p = pos + (size_t)b * NP_ * 3;
  const float qx = pos_s[wave * 3 + 0];
  const float qy = pos_s[wave * 3 + 1];
  const float qz = pos_s[wave * 3 + 2];

  float bv[KNN_]; int bi[KNN_];
#pragma unroll
  for (int k = 0; k < KNN_; ++k) { bv[k] = 3.4e38f; bi[k] = 0x7fffffff; }

  for (int j0 = 0; j0 < NP_; j0 += 32) {
    const int j = j0 + lane;
    const float dx = p[j * 3 + 0] - qx;
    const float dy = p[j * 3 + 1] - qy;
    const float dz = p[j * 3 + 2] - qz;
    const float d2 = dx * dx + dy * dy + dz * dz;
    if (d2 <= R2_ && (d2 < bv[KNN_ - 1] ||
                      (d2 == bv[KNN_ - 1] && j < bi[KNN_ - 1]))) {
      float v = d2; int id = j;
#pragma unroll
      for (int k = 0; k < KNN_; ++k) {
        const bool less = (v < bv[k]) || (v == bv[k] && id < bi[k]);
        const float tv = bv[k]; const int ti = bi[k];
        if (less) { bv[k] = v; bi[k] = id; v = tv; id = ti; }
      }
    }
  }

  int my_j = -1;
#pragma unroll
  for (int k = 0; k < KNN_; ++k) {
    float mv = bv[0]; int mi = bi[0];
    for (int off = 1; off < 32; off <<= 1) {
      const float ov = __shfl_xor(mv, off, 32);
      const int   oi = __shfl_xor(mi, off, 32);
      if (ov < mv || (ov == mv && oi < mi)) { mv = ov; mi = oi; }
    }
    const bool win = (bv[0] == mv) && (bi[0] == mi);
    if (win) {
#pragma unroll
      for (int s = 0; s < KNN_ - 1; ++s) { bv[s] = bv[s + 1]; bi[s] = bi[s + 1]; }
      bv[KNN_ - 1] = 3.4e38f; bi[KNN_ - 1] = 0x7fffffff;
    }
    const int res = (mv < 3.3e38f) ? mi : -1;
    if (lane == k) my_j = res;
  }
  if (lane < KNN_) nbr[wave * KNN_ + lane] = my_j;
}

// =====================================================================
// Fused edge kernel: one wave per sample. Rows of every 16x64 edge matrix
// are the K=16 neighbors -> each matrix is exactly one WMMA A operand.
// =====================================================================
__global__ __launch_bounds__(128) void edge_kernel(
    const float* __restrict__ pos,   const float* __restrict__ pos_s,
    const float* __restrict__ x_proj, const float* __restrict__ q,
    const int* __restrict__ nbr,
    const float* __restrict__ W_pos1, const float* __restrict__ b_pos1,
    const float* __restrict__ W_pos2, const float* __restrict__ b_pos2,
    const float* __restrict__ W_src,  const float* __restrict__ W_attn,
    const float* __restrict__ b_attn, const float* __restrict__ W_lin,
    const float* __restrict__ ln_g,   const float* __restrict__ ln_b,
    float* __restrict__ out)
{
  __shared__ float lds[4][16 * 65];           // per-wave 16x64 scratch, padded
  const int wv   = threadIdx.x >> 5;
  const int lane = threadIdx.x & 31;
  const int i    = blockIdx.x * 4 + wv;       // sample id
  if (i >= B_ * MP_) return;
  const int b    = i / MP_;
  const int m    = lane & 15;                 // A row  AND  B/C column-in-tile
  const int half = lane >> 4;
  float* scratch = lds[wv];

  // my row's neighbor
  const int jm = nbr[i * KNN_ + m];
  const bool valid = jm >= 0;
  const int jj = valid ? jm : 0;
  const float* xrow = x_proj + ((size_t)b * NP_ + jj) * 64;
  const unsigned vmask = (unsigned)(__ballot(valid) & 0xFFFFu);

  const float rx = pos_s[i * 3 + 0] - pos[((size_t)b * NP_ + jj) * 3 + 0];
  const float ry = pos_s[i * 3 + 1] - pos[((size_t)b * NP_ + jj) * 3 + 1];
  const float rz = pos_s[i * 3 + 2] - pos[((size_t)b * NP_ + jj) * 3 + 2];

  // ---- h1 = relu(rel @ W_pos1 + b_pos1)  (K=3, padded to 4) -> scratch
#pragma unroll
  for (int t = 0; t < 4; ++t) {
    const int n0 = t * 16;
    v2f a, bb;
    if (half == 0) { a.x = rx; a.y = ry; bb.x = W_pos1[0 * 64 + n0 + m]; bb.y = W_pos1[1 * 64 + n0 + m]; }
    else           { a.x = rz; a.y = 0.f; bb.x = W_pos1[2 * 64 + n0 + m]; bb.y = 0.f; }
    v8f c = {};
    c = wmma4(a, bb, c);
    const float bn = b_pos1[n0 + m];
#pragma unroll
    for (int r = 0; r < 8; ++r)
      scratch[(r + 8 * half) * 65 + n0 + m] = fmaxf(c[r] + bn, 0.f);
  }

  // ---- delta = relu(h1 @ W_pos2 + b_pos2)   (A-frags from LDS scratch)
  v8f dv[4];
#pragma unroll
  for (int t = 0; t < 4; ++t) {
    const int n0 = t * 16;
    v8f c = {};
    for (int k0 = 0; k0 < 64; k0 += 4) {
      v2f a, bb;
      a.x = scratch[m * 65 + k0 + 2 * half];
      a.y = scratch[m * 65 + k0 + 2 * half + 1];
      bb.x = W_pos2[(k0 + 2 * half) * 64 + n0 + m];
      bb.y = W_pos2[(k0 + 2 * half + 1) * 64 + n0 + m];
      c = wmma4(a, bb, c);
    }
    const float bn = b_pos2[n0 + m];
#pragma unroll
    for (int r = 0; r < 8; ++r) c[r] = fmaxf(c[r] + bn, 0.f);
    dv[t] = c;
  }

  // ---- alpha_in = q_i - x_j@W_src + delta   -> scratch (overwrite)
#pragma unroll
  for (int t = 0; t < 4; ++t) {
    const int n0 = t * 16;
    v8f c = {};
    for (int k0 = 0; k0 < 64; k0 += 4) {
      v2f a, bb;
      a.x = xrow[k0 + 2 * half];
      a.y = xrow[k0 + 2 * half + 1];
      bb.x = W_src[(k0 + 2 * half) * 64 + n0 + m];
      bb.y = W_src[(k0 + 2 * half + 1) * 64 + n0 + m];
      c = wmma4(a, bb, c);
    }
    const float qn = q[(size_t)i * 64 + n0 + m];
#pragma unroll
    for (int r = 0; r < 8; ++r)
      scratch[(r + 8 * half) * 65 + n0 + m] = qn - c[r] + dv[t][r];
  }

  // ---- alpha = relu(alpha_in @ W_attn + b_attn)
  v8f av[4];
#pragma unroll
  for (int t = 0; t < 4; ++t) {
    const int n0 = t * 16;
    v8f c = {};
    for (int k0 = 0; k0 < 64; k0 += 4) {
      v2f a, bb;
      a.x = scratch[m * 65 + k0 + 2 * half];
      a.y = scratch[m * 65 + k0 + 2 * half + 1];
      bb.x = W_attn[(k0 + 2 * half) * 64 + n0 + m];
      bb.y = W_attn[(k0 + 2 * half + 1) * 64 + n0 + m];
      c = wmma4(a, bb, c);
    }
    const float bn = b_attn[n0 + m];
#pragma unroll
    for (int r = 0; r < 8; ++r) c[r] = fmaxf(c[r] + bn, 0.f);
    av[t] = c;
  }

  // ---- masked softmax over the 16 neighbors, per channel ----
#pragma unroll
  for (int t = 0; t < 4; ++t) {
    v8f e = av[t];
    float mx = -3.4e38f;
#pragma unroll
    for (int r = 0; r < 8; ++r)
      if ((vmask >> (r + 8 * half)) & 1) mx = fmaxf(mx, e[r]);
    mx = fmaxf(mx, __shfl_xor(mx, 16, 32));
    float den = 0.f;
#pragma unroll
    for (int r = 0; r < 8; ++r) {
      const float ev = ((vmask >> (r + 8 * half)) & 1) ? __expf(e[r] - mx) : 0.f;
      e[r] = ev; den += ev;
    }
    den += __shfl_xor(den, 16, 32);
    const float inv = 1.0f / den;               // self is always in radius
#pragma unroll
    for (int r = 0; r < 8; ++r) e[r] *= inv;
    av[t] = e;
  }

  // ---- msg = w * (x_j @ W_lin + delta); column (scatter) sums ----
  float o[4];
#pragma unroll
  for (int t = 0; t < 4; ++t) {
    const int n0 = t * 16;
    v8f c = {};
    for (int k0 = 0; k0 < 64; k0 += 4) {
      v2f a, bb;
      a.x = xrow[k0 + 2 * half];
      a.y = xrow[k0 + 2 * half + 1];
      bb.x = W_lin[(k0 + 2 * half) * 64 + n0 + m];
      bb.y = W_lin[(k0 + 2 * half + 1) * 64 + n0 + m];
      c = wmma4(a, bb, c);
    }
    float cs = 0.f;
#pragma unroll
    for (int r = 0; r < 8; ++r) cs += av[t][r] * (c[r] + dv[t][r]);
    cs += __shfl_xor(cs, 16, 32);
    o[t] = cs;                                  // channel t*16 + m (dup halves)
  }

  // ---- LayerNorm over 64 channels ----
  float s1 = o[0] + o[1] + o[2] + o[3];
  float s2 = o[0]*o[0] + o[1]*o[1] + o[2]*o[2] + o[3]*o[3];
  for (int off = 1; off <= 8; off <<= 1) {
    s1 += __shfl_xor(s1, off, 32);
    s2 += __shfl_xor(s2, off, 32);
  }
  const float mean = s1 * (1.0f / 64.0f);
  const float var  = s2 * (1.0f / 64.0f) - mean * mean;
  const float rstd = rsqrtf(var + 1e-5f);
  if (half == 0) {
#pragma unroll
    for (int t = 0; t < 4; ++t) {
      const int n = t * 16 + m;
      out[(size_t)i * 64 + n] = (o[t] - mean) * rstd * ln_g[n] + ln_b[n];
    }
  }
}

// =====================================================================
extern "C" void kernel_launch(void* const* d_in, const int* in_sizes, int n_in,
                              void* d_out, int out_size, void* d_ws, size_t ws_size,
                              hipStream_t stream) {
  const float* x      = (const float*)d_in[0];
  const float* pos    = (const float*)d_in[1];
  // d_in[2]: batch (unused — sorted equal segments)
  const float* W_in   = (const float*)d_in[3];
  const float* b_in   = (const float*)d_in[4];
  const float* W_lin  = (const float*)d_in[5];
  const float* W_src  = (const float*)d_in[6];
  const float* W_dst  = (const float*)d_in[7];
  const float* W_pos1 = (const float*)d_in[8];
  const float* b_pos1 = (const float*)d_in[9];
  const float* W_pos2 = (const float*)d_in[10];
  const float* b_pos2 = (const float*)d_in[11];
  const float* W_attn = (const float*)d_in[12];
  const float* b_attn = (const float*)d_in[13];
  const float* ln_g   = (const float*)d_in[14];
  const float* ln_b   = (const float*)d_in[15];

  float* out     = (float*)d_out;                 // [B*MP, 64]
  float* pos_s   = out + (size_t)B_ * MP_ * 64;   // [B*MP, 3]
  int*   batch_s = (int*)(pos_s + (size_t)B_ * MP_ * 3);

  char* ws = (char*)d_ws;
  float* x_proj = (float*)ws;  ws += (size_t)B_ * NP_ * 64 * sizeof(float);
  float* x_s    = (float*)ws;  ws += (size_t)B_ * MP_ * 64 * sizeof(float);
  float* q      = (float*)ws;  ws += (size_t)B_ * MP_ * 64 * sizeof(float);
  int*   fidx   = (int*)ws;    ws += (size_t)B_ * MP_ * sizeof(int);
  int*   nbr    = (int*)ws;

  // 1) in_proj: x_proj = x @ W_in + b_in
  gemm_n64_kernel<<<(B_ * NP_) / 16, 128, 0, stream>>>(x, W_in, b_in, x_proj, CIN_);
  // 2) farthest point sampling (one block per cloud)
  fps_kernel<<<B_, 1024, 0, stream>>>(pos, fidx);
  // 3) gather sampled pos / features / batch
  gather_kernel<<<(B_ * MP_ * 64) / 256, 256, 0, stream>>>(pos, x_proj, fidx,
                                                           pos_s, x_s, batch_s);
  // 4) radius-kNN (one wave per sample)
  knn_kernel<<<(B_ * MP_) / 4, 128, 0, stream>>>(pos, pos_s, nbr);
  // 5) q = x_s @ W_dst
  gemm_n64_kernel<<<(B_ * MP_) / 16, 128, 0, stream>>>(x_s, W_dst, (const float*)nullptr, q, COUT_);
  // 6) fused edge message + softmax + LayerNorm
  edge_kernel<<<(B_ * MP_) / 4, 128, 0, stream>>>(pos, pos_s, x_proj, q, nbr,
                                                  W_pos1, b_pos1, W_pos2, b_pos2,
                                                  W_src, W_attn, b_attn, W_lin,
                                                  ln_g, ln_b, out);
}